// EarthMoverDistance_31980326486599
// MI455X (gfx1250) — compile-verified
//
#include <hip/hip_runtime.h>

// EMD approxmatch for b=16, n=m=2048 f32 point clouds on gfx1250.
// Cross-term Gram matrix via V_WMMA_F32_16X16X4_F32 (rank-3 GEMM, K padded to 4);
// exp/reduction passes stream from LDS; cost accumulated without materializing match.

#define B_ 16
#define N_ 2048
#define M_ 2048
#define ROWBLK 64              // rows (or cols) per workgroup: 4 waves x 16
#define NBLK (N_ / ROWBLK)     // 32 row-blocks per batch

typedef __attribute__((ext_vector_type(2))) float v2f;
typedef __attribute__((ext_vector_type(8))) float v8f;

// Workspace layout (floats). Total = 6*B*N + B*NBLK = 197120 floats (~788 KB).
enum {
  OFF_REML  = 0,
  OFF_REMR  = B_ * N_,
  OFF_RATL  = 2 * B_ * N_,
  OFF_RATR  = 3 * B_ * N_,
  OFF_N1SQ  = 4 * B_ * N_,
  OFF_N2SQ  = 5 * B_ * N_,
  OFF_COSTP = 6 * B_ * N_,     // B_*NBLK per-(batch,rowblock) cost partials
};

// Build a WMMA 16x4 f32 A-fragment (or 4x16 B-fragment: same per-lane pattern)
// for point idx of a packed float3 array.
//   lanes 0-15 : (K0,K1) = (x, y)
//   lanes16-31 : (K2,K3) = (z, 0)
__device__ __forceinline__ v2f make_frag(const float* xyz, int idx, bool hi) {
  v2f f;
  f.x = xyz[idx * 3 + (hi ? 2 : 0)];
  f.y = hi ? 0.0f : xyz[idx * 3 + 1];
  return f;
}

__device__ __forceinline__ v8f cross16(v2f a, v2f b) {
  v8f c = {};
  // D = A(16x4) x B(4x16) + 0   -> per-element dot(x1_i, x2_j)
  return __builtin_amdgcn_wmma_f32_16x16x4_f32(
      /*neg_a=*/false, a, /*neg_b=*/false, b,
      /*c_mod=*/(short)0, c, /*reuse_a=*/false, /*reuse_b=*/false);
}

__global__ void emd_init(const float* __restrict__ x1,
                         const float* __restrict__ x2,
                         float* __restrict__ ws) {
  int i = blockIdx.x * blockDim.x + threadIdx.x;   // over B_*N_
  if (i < B_ * N_) {
    const float* p = x1 + i * 3;
    const float* q = x2 + i * 3;
    ws[OFF_N1SQ + i] = p[0] * p[0] + p[1] * p[1] + p[2] * p[2];
    ws[OFF_N2SQ + i] = q[0] * q[0] + q[1] * q[1] + q[2] * q[2];
    ws[OFF_REML + i] = 1.0f;   // n == m  ->  multiL = multiR = 1
    ws[OFF_REMR + i] = 1.0f;
  }
  if (i < B_ * NBLK) ws[OFF_COSTP + i] = 0.0f;
}

// Row pass. MODE 0: t1 = A @ remainR; ratioL = remainL / (1e-9 + t1).
//           MODE 1: s = A @ ratioR, c = (A*d) @ ratioR;
//                   remainL = max(remainL - ratioL*s, 0); cost += ratioL*c.
template <int MODE>
__global__ __launch_bounds__(128) void emd_rowpass(const float* __restrict__ x1,
                                                   const float* __restrict__ x2,
                                                   float* __restrict__ ws,
                                                   float level) {
  __shared__ float s_xyz[M_ * 3];   // 24 KB: batch's xyz2 cloud
  __shared__ float s_nsq[M_];       //  8 KB: |x2|^2
  __shared__ float s_vec[M_];       //  8 KB: remainR (MODE 0) / ratioR (MODE 1)
  __shared__ float s_cost[8];

  const int b   = blockIdx.x;
  const int tid = threadIdx.x;

  const float* x2b  = x2 + (size_t)b * M_ * 3;
  const float* n2sq = ws + OFF_N2SQ + b * M_;
  const float* vecg = ws + (MODE == 0 ? OFF_REMR : OFF_RATR) + b * M_;
  for (int i = tid; i < M_ * 3; i += 128) s_xyz[i] = x2b[i];
  for (int i = tid; i < M_; i += 128) { s_nsq[i] = n2sq[i]; s_vec[i] = vecg[i]; }
  __syncthreads();

  const int  wave = tid >> 5;
  const int  lane = tid & 31;
  const bool hi   = lane >= 16;
  const int  l15  = lane & 15;
  const int  rowbase = blockIdx.y * ROWBLK + wave * 16;

  const float* x1b = x1 + (size_t)b * N_ * 3;
  const v2f af = make_frag(x1b, rowbase + l15, hi);   // invariant over m-loop

  const float* n1sq = ws + OFF_N1SQ + b * N_;
  float rsq[8];
#pragma unroll
  for (int r = 0; r < 8; ++r) rsq[r] = n1sq[rowbase + r + (hi ? 8 : 0)];

  float accS[8], accC[8];
#pragma unroll
  for (int r = 0; r < 8; ++r) { accS[r] = 0.0f; accC[r] = 0.0f; }

  for (int ct = 0; ct < M_ / 16; ++ct) {
    const int c = ct * 16 + l15;
    const v2f bf = make_frag(s_xyz, c, hi);
    const v8f cr = cross16(af, bf);
    const float csq = s_nsq[c];
    const float rv  = s_vec[c];
#pragma unroll
    for (int r = 0; r < 8; ++r) {
      // D element (VGPR r, lane): row = rowbase + r + 8*hi, col = lane&15
      const float d2 = rsq[r] + csq - 2.0f * cr[r];
      const float a  = __expf(level * d2) * rv;
      accS[r] += a;
      if (MODE == 1) accC[r] += a * sqrtf(fmaxf(d2, 1e-40f));
    }
  }

  // Reduce each row accumulator across its 16-lane half.
#pragma unroll
  for (int r = 0; r < 8; ++r) {
    float v = accS[r];
    v += __shfl_xor(v, 1); v += __shfl_xor(v, 2);
    v += __shfl_xor(v, 4); v += __shfl_xor(v, 8);
    accS[r] = v;
    if (MODE == 1) {
      float w = accC[r];
      w += __shfl_xor(w, 1); w += __shfl_xor(w, 2);
      w += __shfl_xor(w, 4); w += __shfl_xor(w, 8);
      accC[r] = w;
    }
  }

  float mycost = 0.0f;
  if (l15 == 0) {   // lane 0 owns rows rowbase+0..7, lane 16 owns rowbase+8..15
    float* remL = ws + OFF_REML + b * N_;
    float* ratL = ws + OFF_RATL + b * N_;
#pragma unroll
    for (int r = 0; r < 8; ++r) {
      const int row = rowbase + r + (hi ? 8 : 0);
      if (MODE == 0) {
        ratL[row] = remL[row] / (1e-9f + accS[r]);
      } else {
        const float rl = ratL[row];
        remL[row] = fmaxf(remL[row] - rl * accS[r], 0.0f);
        mycost += rl * accC[r];
      }
    }
  }

  if (MODE == 1) {
    if (l15 == 0) s_cost[wave * 2 + (hi ? 1 : 0)] = mycost;
    __syncthreads();
    if (tid == 0) {
      float t = 0.0f;
#pragma unroll
      for (int i = 0; i < 8; ++i) t += s_cost[i];
      ws[OFF_COSTP + b * NBLK + blockIdx.y] += t;   // WG-exclusive slot: deterministic
    }
  }
}

// Column pass: t2 = A^T @ ratioL; ratioR = min(remainR/(remainR*t2 + 1e-9), 1)*remainR;
// remainR = max(remainR - remainR*t2, 0).
__global__ __launch_bounds__(128) void emd_colpass(const float* __restrict__ x1,
                                                   const float* __restrict__ x2,
                                                   float* __restrict__ ws,
                                                   float level) {
  __shared__ float s_xyz[N_ * 3];
  __shared__ float s_nsq[N_];
  __shared__ float s_rl[N_];

  const int b   = blockIdx.x;
  const int tid = threadIdx.x;

  const float* x1b  = x1 + (size_t)b * N_ * 3;
  const float* n1sq = ws + OFF_N1SQ + b * N_;
  const float* ratL = ws + OFF_RATL + b * N_;
  for (int i = tid; i < N_ * 3; i += 128) s_xyz[i] = x1b[i];
  for (int i = tid; i < N_; i += 128) { s_nsq[i] = n1sq[i]; s_rl[i] = ratL[i]; }
  __syncthreads();

  const int  wave = tid >> 5;
  const int  lane = tid & 31;
  const bool hi   = lane >= 16;
  const int  l15  = lane & 15;
  const int  colbase = blockIdx.y * ROWBLK + wave * 16;

  const float* x2b = x2 + (size_t)b * M_ * 3;
  const v2f bf = make_frag(x2b, colbase + l15, hi);            // invariant
  const float csq = (ws + OFF_N2SQ + b * M_)[colbase + l15];

  float acc = 0.0f;   // column partial for col = colbase + l15 (this half's rows)
  for (int rt = 0; rt < N_ / 16; ++rt) {
    const v2f af = make_frag(s_xyz, rt * 16 + l15, hi);
    const v8f cr = cross16(af, bf);
    const int rb = rt * 16 + (hi ? 8 : 0);
#pragma unroll
    for (int r = 0; r < 8; ++r) {
      const int row = rb + r;                                   // broadcast LDS reads
      const float d2 = s_nsq[row] + csq - 2.0f * cr[r];
      acc += __expf(level * d2) * s_rl[row];
    }
  }
  acc += __shfl_xor(acc, 16);   // combine the two row-halves

  if (!hi) {
    const int col = colbase + l15;
    float* remR = ws + OFF_REMR + b * M_;
    float* ratR = ws + OFF_RATR + b * M_;
    const float rr   = remR[col];
    const float sumr = rr * acc;
    ratR[col] = fminf(rr / (sumr + 1e-9f), 1.0f) * rr;
    remR[col] = fmaxf(rr - sumr, 0.0f);
  }
}

__global__ void emd_finalize(const float* __restrict__ ws, float* __restrict__ out) {
  if (blockIdx.x == 0 && threadIdx.x == 0) {
    float tot = 0.0f;
    for (int i = 0; i < B_ * NBLK; ++i) tot += ws[OFF_COSTP + i];  // fixed order
    out[0] = tot / (float)B_;   // mean over batch
  }
}

extern "C" void kernel_launch(void* const* d_in, const int* in_sizes, int n_in,
                              void* d_out, int out_size, void* d_ws, size_t ws_size,
                              hipStream_t stream) {
  (void)in_sizes; (void)n_in; (void)out_size; (void)ws_size;
  const float* x1 = (const float*)d_in[0];
  const float* x2 = (const float*)d_in[1];
  float* ws  = (float*)d_ws;
  float* out = (float*)d_out;

  static const float levels[10] = {
      -16384.0f, -4096.0f, -1024.0f, -256.0f, -64.0f,
      -16.0f, -4.0f, -1.0f, -0.25f, 0.0f};

  emd_init<<<dim3((B_ * N_ + 255) / 256), 256, 0, stream>>>(x1, x2, ws);

  for (int l = 0; l < 10; ++l) {
    emd_rowpass<0><<<dim3(B_, NBLK), 128, 0, stream>>>(x1, x2, ws, levels[l]);
    emd_colpass   <<<dim3(B_, NBLK), 128, 0, stream>>>(x1, x2, ws, levels[l]);
    emd_rowpass<1><<<dim3(B_, NBLK), 128, 0, stream>>>(x1, x2, ws, levels[l]);
  }

  emd_finalize<<<1, 1, 0, stream>>>(ws, out);
}